// Nn2_54254026883600
// MI455X (gfx1250) — compile-verified
//
#include <hip/hip_runtime.h>
#include <hip/hip_bf16.h>

#define DIMN   128
#define DNUM   6
#define NTOK   50257
#define INDIM  131072      // 512*256
#define VCOLS  768         // DIM*D
#define VCOLS4 192         // 768/4

#define NXB    96          // blocks for x0 partials
#define NKB    512         // blocks for projV GEMV
#define KCH    256         // rows per block (512*256 = 131072)

typedef __attribute__((ext_vector_type(2))) float v2f;
typedef __attribute__((ext_vector_type(8))) float v8f;

// ---------------- Kernel 1: x0 partials: t0 @ projX0 (131072 x 6) ----------
__global__ __launch_bounds__(256) void k_x0_part(const float* __restrict__ t0,
                                                 const float* __restrict__ PX,
                                                 float* __restrict__ part) {
    __shared__ float red[256];
    float acc[DNUM] = {0.f, 0.f, 0.f, 0.f, 0.f, 0.f};
    for (int k = blockIdx.x * 256 + threadIdx.x; k < INDIM; k += NXB * 256) {
        float t = t0[k];
        const float* row = PX + (size_t)k * DNUM;
#pragma unroll
        for (int j = 0; j < DNUM; ++j) acc[j] += t * row[j];
    }
#pragma unroll
    for (int j = 0; j < DNUM; ++j) {
        red[threadIdx.x] = acc[j];
        __syncthreads();
        for (int s = 128; s > 0; s >>= 1) {
            if (threadIdx.x < (unsigned)s) red[threadIdx.x] += red[threadIdx.x + s];
            __syncthreads();
        }
        if (threadIdx.x == 0) part[blockIdx.x * DNUM + j] = red[0];
        __syncthreads();
    }
}

// ------- Kernel 2: projV GEMV partials. Thread owns one float4 column group.
__global__ __launch_bounds__(192) void k_v_part(const float* __restrict__ t0,
                                                const float4* __restrict__ PV4,
                                                float* __restrict__ part) {
    __shared__ float s_t0[KCH];
    const int tid = threadIdx.x;
    const int k0  = blockIdx.x * KCH;
    for (int i = tid; i < KCH; i += 192) s_t0[i] = t0[k0 + i];
    __syncthreads();

    float4 acc0 = {0.f, 0.f, 0.f, 0.f};
    float4 acc1 = {0.f, 0.f, 0.f, 0.f};
    const float4* row = PV4 + (size_t)k0 * VCOLS4 + tid;
#pragma unroll 4
    for (int r = 0; r < KCH; r += 2) {
        float4 p0 = row[(size_t)r * VCOLS4];
        float4 p1 = row[(size_t)(r + 1) * VCOLS4];
        float s0 = s_t0[r], s1 = s_t0[r + 1];
        acc0.x += s0 * p0.x; acc0.y += s0 * p0.y; acc0.z += s0 * p0.z; acc0.w += s0 * p0.w;
        acc1.x += s1 * p1.x; acc1.y += s1 * p1.y; acc1.z += s1 * p1.z; acc1.w += s1 * p1.w;
    }
    float* dst = part + (size_t)blockIdx.x * VCOLS + tid * 4;
    dst[0] = acc0.x + acc1.x;
    dst[1] = acc0.y + acc1.y;
    dst[2] = acc0.z + acc1.z;
    dst[3] = acc0.w + acc1.w;
}

// ---------------- Kernel 3: deterministic reduction of v partials ----------
__global__ __launch_bounds__(256) void k_v_reduce(const float* __restrict__ part,
                                                  float* __restrict__ v) {
    int c = blockIdx.x * 256 + threadIdx.x;
    if (c >= VCOLS) return;
    float s = 0.f;
    for (int b = 0; b < NKB; ++b) s += part[(size_t)b * VCOLS + c];
    v[c] = s;
}

// -------- Kernel 4: x0 reduce + filter/Toeplitz weights + t = w . v --------
__global__ __launch_bounds__(128) void k_mid(const float* __restrict__ x0part,
                                             const float* __restrict__ a,
                                             const float* __restrict__ d0,
                                             const float* __restrict__ v,
                                             float* __restrict__ tout) {
    __shared__ float sx0[DNUM];
    __shared__ float sw[DIMN];
    const int tid = threadIdx.x;
    if (tid < DNUM) {
        float s = 0.f;
        for (int b = 0; b < NXB; ++b) s += x0part[b * DNUM + tid];
        sx0[tid] = s;
    }
    __syncthreads();
    {
        const int i = tid;   // 0..127
        float s = 0.f;
#pragma unroll
        for (int d = 0; d < DNUM; ++d) {
            int j = i - d;
            if (j >= 0) {
                float filt = expf(-0.3f * (float)j) * (1.0f - 0.0005f) + 0.0005f;
                s += sx0[d] * a[j] * filt;
            }
        }
        sw[i] = d0[i] * s;
    }
    __syncthreads();
    if (tid < DNUM) {
        float s = 0.f;
        for (int i = 0; i < DIMN; ++i) s += sw[i] * v[tid * DIMN + i];
        tout[tid] = s;
    }
}

// -------- Kernel 5: out = t(1x6) @ W_out(6xNTOK) + b_out via WMMA f32 ------
// Per wave: one 16-column tile. A(16x4) has all 16 rows == t, so every row of
// D equals the result. K=6 split across two 16x16x4 steps (rows 6,7 zeroed).
__global__ __launch_bounds__(128) void k_proj_wmma(const float* __restrict__ t6,
                                                   const float* __restrict__ W,
                                                   const float* __restrict__ bias,
                                                   float* __restrict__ out) {
    const int lane = threadIdx.x & 31;
    const int wave = blockIdx.x * 4 + (threadIdx.x >> 5);
    const int n0   = wave * 16;

    float tv[DNUM];
#pragma unroll
    for (int d = 0; d < DNUM; ++d) tv[d] = t6[d];

    const int  nloc = lane & 15;
    const bool hi   = lane >= 16;
    const int  n    = n0 + nloc;
    const int  nc   = n < NTOK ? n : (NTOK - 1);   // clamp loads; EXEC stays full

    // Step 0: K rows 0..3.  A vgpr0: K=0 (lanes 0-15) / K=2 (16-31); vgpr1: K=1 / K=3.
    v2f a0, b0, a1, b1;
    a0.x = hi ? tv[2] : tv[0];
    a0.y = hi ? tv[3] : tv[1];
    b0.x = W[(size_t)(hi ? 2 : 0) * NTOK + nc];
    b0.y = W[(size_t)(hi ? 3 : 1) * NTOK + nc];
    // Step 1: K rows 4..7 (rows 6,7 are zero padding).
    a1.x = hi ? 0.f : tv[4];
    a1.y = hi ? 0.f : tv[5];
    b1.x = hi ? 0.f : W[(size_t)4 * NTOK + nc];
    b1.y = hi ? 0.f : W[(size_t)5 * NTOK + nc];

    v8f c = {};
    c = __builtin_amdgcn_wmma_f32_16x16x4_f32(false, a0, false, b0, (short)0, c, false, false);
    c = __builtin_amdgcn_wmma_f32_16x16x4_f32(false, a1, false, b1, (short)0, c, false, false);

    // D row 0 lives in c[0] on lanes 0..15 (N = lane).
    if (!hi && n < NTOK) out[n] = c[0] + bias[n];
}

extern "C" void kernel_launch(void* const* d_in, const int* in_sizes, int n_in,
                              void* d_out, int out_size, void* d_ws, size_t ws_size,
                              hipStream_t stream) {
    const float* xIn    = (const float*)d_in[0];   // 512*256, flat == t0
    const float* projX0 = (const float*)d_in[1];   // 131072 x 6
    const float* projV  = (const float*)d_in[2];   // 131072 x 768
    const float* a      = (const float*)d_in[3];   // 128
    const float* d0     = (const float*)d_in[4];   // 128
    const float* W_out  = (const float*)d_in[5];   // 6 x 50257
    const float* b_out  = (const float*)d_in[6];   // 50257
    float* out = (float*)d_out;

    float* ws     = (float*)d_ws;
    float* x0part = ws;                                   // NXB*6      = 576
    float* vpart  = x0part + NXB * DNUM;                  // NKB*768    = 393216
    float* vred   = vpart + (size_t)NKB * VCOLS;          // 768
    float* t6     = vred + VCOLS;                         // 6
    (void)in_sizes; (void)n_in; (void)out_size; (void)ws_size;

    k_x0_part <<<NXB, 256, 0, stream>>>(xIn, projX0, x0part);
    k_v_part  <<<NKB, 192, 0, stream>>>(xIn, (const float4*)projV, vpart);
    k_v_reduce<<<3, 256, 0, stream>>>(vpart, vred);
    k_mid     <<<1, 128, 0, stream>>>(x0part, a, d0, vred, t6);
    k_proj_wmma<<<(NTOK + 63) / 64, 128, 0, stream>>>(t6, W_out, b_out, out);
}